// AFM_56392920596760
// MI455X (gfx1250) — compile-verified
//
#include <hip/hip_runtime.h>
#include <hip/hip_bf16.h>

#define F_FIELDS 32
#define E_DIM    64

typedef float v2f __attribute__((ext_vector_type(2)));
typedef float v8f __attribute__((ext_vector_type(8)));

// ---------------------------------------------------------------------------
// Kernel 1: gather-bound phase. One wave32 per sample.
//  first[b]  = sum_i e1[i, x[b,i]]
//  second[b] = sum_i e2[i, x[b,i]] * (sum_j e2[i, x[b,j]])   (elementwise over E)
// Row addresses are wave-uniform (broadcast via v_readlane -> saddr loads);
// lanes stride float2 across E=64 -> each gather is one coalesced 256B row.
// ---------------------------------------------------------------------------
__global__ __launch_bounds__(256) void afm_gather_kernel(
    const int* __restrict__ x, const float* __restrict__ e1,
    const float* __restrict__ e2, float* __restrict__ firstv,
    float* __restrict__ second, int V, int B) {
  const int lane = threadIdx.x & 31;
  const int wave = threadIdx.x >> 5;
  const long long b = (long long)blockIdx.x * (blockDim.x >> 5) + wave;
  if (b >= B) return;  // wave-uniform

  const int xi = x[b * F_FIELDS + lane];  // lane j holds x[b,j]

  // ---- first order: lane i gathers e1[i, x[b,i]], then wave reduction ----
  float fv = e1[(size_t)lane * V + xi];
#pragma unroll
  for (int off = 16; off > 0; off >>= 1) fv += __shfl_xor(fv, off, 32);
  if (lane == 0) firstv[b] = fv;

  // ---- second order ----
  float2 acc = make_float2(0.f, 0.f);
  for (int i = 0; i < F_FIELDS; ++i) {
    const float* tbl = e2 + (size_t)i * V * E_DIM;
    float2 s = make_float2(0.f, 0.f);
    float2 d = make_float2(0.f, 0.f);
#pragma unroll 8
    for (int j = 0; j < F_FIELDS; ++j) {
      const int idx = __builtin_amdgcn_readlane(xi, j);  // SGPR index
      const float2 r =
          *(const float2*)(tbl + (size_t)idx * E_DIM + 2 * lane);
      s.x += r.x;
      s.y += r.y;
      if (j == i) d = r;  // d_i = e2[i, x[b,i]]
    }
    acc.x += d.x * s.x;
    acc.y += d.y * s.y;
  }
  ((float2*)(second + (size_t)b * E_DIM))[lane] = acc;
}

// ---------------------------------------------------------------------------
// Kernel 2: attention via WMMA f32 16x16x4 + softmax + output head.
// 128 threads (4 waves) per block; block handles 16 samples.
// Wave w computes the 16x16 logits tile for columns [16w,16w+16),
// accumulating over K=64 in 16 steps of K=4.
// ---------------------------------------------------------------------------
__global__ __launch_bounds__(128) void afm_attn_kernel(
    const float* __restrict__ second, const float* __restrict__ firstv,
    const float* __restrict__ W_att, const float* __restrict__ W_out,
    const float* __restrict__ b_out, float* __restrict__ out) {
  __shared__ float s_sec[16][E_DIM];  // second tile (also reused for pooled)
  __shared__ float s_log[16][E_DIM];  // logits -> exp values

  const int t = threadIdx.x;
  // Stage the 16x64 'second' tile into LDS (1024 floats, float4 loads).
  {
    const float4* src =
        (const float4*)(second + (size_t)blockIdx.x * 16 * E_DIM);
    float4* dst = (float4*)&s_sec[0][0];
    dst[t] = src[t];
    dst[t + 128] = src[t + 128];
  }
  __syncthreads();

  const int lane = t & 31;
  const int w = t >> 5;       // N-tile index 0..3
  const int half = lane >> 4; // K sub-split per ISA A/B frag layout
  const int m = lane & 15;

  v8f c = {};
#pragma unroll
  for (int k0 = 0; k0 < E_DIM; k0 += 4) {
    const int ka = k0 + 2 * half;
    v2f a, bb;
    // A 16x4 frag: v0 = A[m, k0 | k0+2], v1 = A[m, k0+1 | k0+3]
    a.x = s_sec[m][ka];
    a.y = s_sec[m][ka + 1];
    // B 4x16 frag (row striped across lanes), W_att[k, n] row-major
    bb.x = W_att[ka * E_DIM + 16 * w + m];
    bb.y = W_att[(ka + 1) * E_DIM + 16 * w + m];
    c = __builtin_amdgcn_wmma_f32_16x16x4_f32(
        /*neg_a=*/false, a, /*neg_b=*/false, bb,
        /*c_mod=*/(short)0, c, /*reuse_a=*/false, /*reuse_b=*/false);
  }
  // C/D layout: VGPR r -> row r (lanes 0-15) / row r+8 (lanes 16-31), col=m
#pragma unroll
  for (int r = 0; r < 8; ++r) {
    const int row = half ? (r + 8) : r;
    s_log[row][16 * w + m] = c[r];
  }
  __syncthreads();

  // Softmax over E + pooled dot with W_out; 16 threads, one row each (tiny).
  if (t < 16) {
    const int row = t;
    const size_t b = (size_t)blockIdx.x * 16 + row;
    float mx = -__builtin_inff();
    for (int e = 0; e < E_DIM; ++e) mx = fmaxf(mx, s_log[row][e]);
    float ssum = 0.f;
    for (int e = 0; e < E_DIM; ++e) {
      const float ex = __expf(s_log[row][e] - mx);
      s_log[row][e] = ex;
      ssum += ex;
    }
    const float inv = 1.f / ssum;
    float acc = firstv[b] * W_out[0] + b_out[0];
    for (int e = 0; e < E_DIM; ++e)
      acc += (s_log[row][e] * inv) * s_sec[row][e] * W_out[1 + e];
    out[b] = acc;
  }
}

// ---------------------------------------------------------------------------
extern "C" void kernel_launch(void* const* d_in, const int* in_sizes, int n_in,
                              void* d_out, int out_size, void* d_ws,
                              size_t ws_size, hipStream_t stream) {
  const int* x = (const int*)d_in[0];          // (B, 32)
  const float* e1 = (const float*)d_in[1];     // (32, V)
  const float* e2 = (const float*)d_in[2];     // (32, V, 64)
  const float* W_att = (const float*)d_in[3];  // (64, 64)
  const float* W_out = (const float*)d_in[4];  // (65, 1)
  const float* b_out = (const float*)d_in[5];  // (1,)
  float* out = (float*)d_out;                  // (B, 1)

  const int B = in_sizes[0] / F_FIELDS;  // 16384
  const int V = in_sizes[1] / F_FIELDS;  // 100000

  // Workspace: [first: B floats][second: B*64 floats]
  float* firstv = (float*)d_ws;
  float* second = firstv + B;

  const int waves_per_block = 8;  // 256 threads
  dim3 blk1(32 * waves_per_block);
  dim3 grd1((B + waves_per_block - 1) / waves_per_block);
  afm_gather_kernel<<<grd1, blk1, 0, stream>>>(x, e1, e2, firstv, second, V, B);

  dim3 blk2(128);
  dim3 grd2(B / 16);
  afm_attn_kernel<<<grd2, blk2, 0, stream>>>(second, firstv, W_att, W_out,
                                             b_out, out);
}